// RoIAlign_34737695490233
// MI455X (gfx1250) — compile-verified
//
#include <hip/hip_runtime.h>
#include <math.h>

// Problem constants (from reference)
#define BB   8
#define CC   256
#define HH   200
#define WW   200
#define SS   7
#define RR   2048          // B*N rois
#define GK   196           // 14*14 grid points = 49 K-steps of 4
#define MROWS 64           // 49 output positions padded to 64 (4 M-tiles of 16)

typedef float v2f __attribute__((ext_vector_type(2)));
typedef float v8f __attribute__((ext_vector_type(8)));

__global__ __launch_bounds__(256)
void roialign_wmma_kernel(const float* __restrict__ fm,
                          const int*   __restrict__ props,
                          float*       __restrict__ out) {
    __shared__ int   s_yy[14];
    __shared__ int   s_xx[14];
    __shared__ float s_wy[SS];
    __shared__ float s_wx[SS];
    __shared__ int   s_off[GK];
    __shared__ float s_W[MROWS][GK];   // 50 KB: per-ROI sparse weight matrix

    const int r     = blockIdx.x;       // roi index
    const int tid   = threadIdx.x;
    const int batch = r >> 8;           // r / N, N == 256

    // ---- Phase 0: box + per-axis sampling (7 threads) ----
    if (tid < SS) {
        int x1 = props[r * 4 + 0];
        int y1 = props[r * 4 + 1];
        int x2 = props[r * 4 + 2];
        int y2 = props[r * 4 + 3];
        int bx1 = min(max(x1, 0), WW - 1);
        int by1 = min(max(y1, 0), HH - 1);
        int bx2 = min(max(x2, 0), WW - 1);
        int by2 = min(max(y2, 0), HH - 1);
        int h = by2 - by1 + 1;
        int w = bx2 - bx1 + 1;

        float gy = ((float)tid + 0.5f) * ((float)h / (float)SS) - 0.5f;
        float srcy = fmaxf(gy, 0.0f);
        int   y0   = (int)floorf(srcy);
        float wy   = srcy - (float)y0;
        int   y1i  = min(y0 + 1, h - 1);
        s_yy[2 * tid]     = by1 + y0;
        s_yy[2 * tid + 1] = by1 + y1i;
        s_wy[tid]         = wy;

        float gx = ((float)tid + 0.5f) * ((float)w / (float)SS) - 0.5f;
        float srcx = fmaxf(gx, 0.0f);
        int   x0   = (int)floorf(srcx);
        float wx   = srcx - (float)x0;
        int   x1i  = min(x0 + 1, w - 1);
        s_xx[2 * tid]     = bx1 + x0;
        s_xx[2 * tid + 1] = bx1 + x1i;
        s_wx[tid]         = wx;
    }
    __syncthreads();

    // ---- Phase 1: grid-point offsets within one (b,c) plane ----
    if (tid < GK) {
        int j = tid / 14;
        int i = tid - j * 14;
        s_off[tid] = s_yy[j] * WW + s_xx[i];
    }

    // ---- Phase 2: dense weight matrix W[64][196] (rows >=49 are zero) ----
    // 64*196 = 12544 = 49 entries per thread, computed analytically (no scatter)
    for (int t = 0; t < 49; ++t) {
        int idx = tid + t * 256;
        int m = idx / GK;
        int g = idx - m * GK;
        float val = 0.0f;
        if (m < 49) {
            int syp = m / SS, sxp = m - syp * SS;
            int j = g / 14, i = g - j * 14;
            float fy = 0.0f, fx = 0.0f;
            if ((j >> 1) == syp) fy = (j & 1) ? s_wy[syp] : (1.0f - s_wy[syp]);
            if ((i >> 1) == sxp) fx = (i & 1) ? s_wx[sxp] : (1.0f - s_wx[sxp]);
            val = fy * fx;
        }
        s_W[m][g] = val;
    }
    __syncthreads();

    // ---- Phase 3: per-wave GEMM  D(64x16) = W(64x196) x G(196x16) ----
    const int wave   = tid >> 5;
    const int lane   = tid & 31;
    const int nloc   = lane & 15;                 // N (channel-in-block)
    const int khalf  = (lane < 16) ? 0 : 2;      // lanes 0-15: K=0,1  lanes 16-31: K=2,3
    const int rowadd = (lane < 16) ? 0 : 8;      // D layout: M = v (+8 for hi lanes)

    for (int cbi = 0; cbi < 2; ++cbi) {
        const int cblock = wave + cbi * 8;                       // 0..15
        const int c      = cblock * 16 + nloc;
        const float* __restrict__ fmc =
            fm + (size_t)(batch * CC + c) * (size_t)(HH * WW);

        v8f acc0 = {}, acc1 = {}, acc2 = {}, acc3 = {};

        for (int kk = 0; kk < 49; ++kk) {
            const int g0 = kk * 4 + khalf;
            const int off0 = s_off[g0];
            const int off1 = s_off[g0 + 1];

            v2f b;
            b[0] = fmc[off0];
            b[1] = fmc[off1];

            if (kk + 1 < 49) {
                // gfx1250 global_prefetch_b8 for next K-step
                __builtin_prefetch(&fmc[s_off[g0 + 4]], 0, 1);
            }

            v2f a0, a1, a2, a3;
            a0[0] = s_W[nloc][g0];        a0[1] = s_W[nloc][g0 + 1];
            a1[0] = s_W[16 + nloc][g0];   a1[1] = s_W[16 + nloc][g0 + 1];
            a2[0] = s_W[32 + nloc][g0];   a2[1] = s_W[32 + nloc][g0 + 1];
            a3[0] = s_W[48 + nloc][g0];   a3[1] = s_W[48 + nloc][g0 + 1];

            acc0 = __builtin_amdgcn_wmma_f32_16x16x4_f32(
                false, a0, false, b, (short)0, acc0, false, false);
            acc1 = __builtin_amdgcn_wmma_f32_16x16x4_f32(
                false, a1, false, b, (short)0, acc1, false, false);
            acc2 = __builtin_amdgcn_wmma_f32_16x16x4_f32(
                false, a2, false, b, (short)0, acc2, false, false);
            acc3 = __builtin_amdgcn_wmma_f32_16x16x4_f32(
                false, a3, false, b, (short)0, acc3, false, false);
        }

        // ---- store D: out[r][c][p] with p = tile*16 + v + rowadd ----
        float* __restrict__ outc = out + ((size_t)r * CC + c) * 49;
#pragma unroll
        for (int v = 0; v < 8; ++v) {
            const int pv = v + rowadd;
            outc[pv]      = acc0[v];
            outc[16 + pv] = acc1[v];
            outc[32 + pv] = acc2[v];
            if (48 + pv < 49) outc[48 + pv] = acc3[v];
        }
    }
}

extern "C" void kernel_launch(void* const* d_in, const int* in_sizes, int n_in,
                              void* d_out, int out_size, void* d_ws, size_t ws_size,
                              hipStream_t stream) {
    const float* fm    = (const float*)d_in[0];   // (8,256,200,200) fp32
    const int*   props = (const int*)d_in[1];     // (8,256,4) int32
    float*       out   = (float*)d_out;           // (2048,256,7,7) fp32
    (void)in_sizes; (void)n_in; (void)out_size; (void)d_ws; (void)ws_size;

    dim3 grid(RR);     // one workgroup per ROI
    dim3 block(256);   // 8 waves; each wave owns two 16-channel blocks
    hipLaunchKernelGGL(roialign_wmma_kernel, grid, block, 0, stream,
                       fm, props, out);
}